// TSBarrierModel_37091337568669
// MI455X (gfx1250) — compile-verified
//
#include <hip/hip_runtime.h>
#include <hip/hip_bf16.h>
#include <math.h>

typedef float v2f __attribute__((ext_vector_type(2)));
typedef float v4f __attribute__((ext_vector_type(4)));
typedef float v8f __attribute__((ext_vector_type(8)));

#define WAVES_PER_BLOCK 8
#define BLOCK_THREADS 256
#define EMB_PITCH 12   // floats per emb row (48 B, 16B-aligned for b128 stores)
#define H_PITCH   20   // floats per h row (80 B, keeps b64 reads conflict-free)

// Each wave processes one 32-edge tile per iteration (1 edge per lane).
// GEMM1 (per 16-edge half): emb[16x12] @ W1[12x16]          -> 3x v_wmma_f32_16x16x4_f32
// GEMM2 (per 16-edge half): h[16x16] @ (W2*PATH_C/4)[16x16] -> 4x v_wmma_f32_16x16x4_f32
__global__ void __launch_bounds__(BLOCK_THREADS)
tsb_edge_kernel(const float* __restrict__ nf,   // [N,9]
                const float* __restrict__ ev,   // [E,3]
                const float* __restrict__ W1,   // [10,16]
                const float* __restrict__ W2,   // [16,3]
                const int*   __restrict__ esrc, // [E]
                float* __restrict__ partials,   // [gridDim.x]
                int E, int tiles, int n_iter, int stride)
{
    __shared__ __align__(16) float lds_emb[WAVES_PER_BLOCK][32 * EMB_PITCH];
    __shared__ __align__(16) float lds_h[WAVES_PER_BLOCK][2][16 * H_PITCH];
    __shared__ __align__(16) float lds_d[WAVES_PER_BLOCK][32 * 4];
    __shared__ float red[WAVES_PER_BLOCK];

    const int tid  = threadIdx.x;
    const int lane = tid & 31;
    const int wv   = tid >> 5;
    const int lo   = lane & 15;
    const int hi   = lane >> 4;                 // 0: lanes 0-15, 1: lanes 16-31
    const int wid  = blockIdx.x * WAVES_PER_BLOCK + wv;

    // ---- loop-invariant B operands -------------------------------------
    // Assumed 4x16 f32 B layout: VGPR0 -> rows {4c+0 (lanes0-15), 4c+2 (lanes16-31)},
    //                            VGPR1 -> rows {4c+1, 4c+3}; column = lane%16.
    v2f b1[3];
#pragma unroll
    for (int c = 0; c < 3; ++c) {
        const int k0 = 4 * c + (hi ? 2 : 0);
        const int k1 = 4 * c + (hi ? 3 : 1);
        b1[c].x = (k0 < 10) ? W1[k0 * 16 + lo] : 0.0f;
        b1[c].y = (k1 < 10) ? W1[k1 * 16 + lo] : 0.0f;
    }
    // PATH_C[o]/sqrt(HIDDEN) folded into B2; columns >=3 are zero.
    const float PC0 = 0.5773502691896258f;   // 1/sqrt(3)
    const float PC1 = 0.3333333333333333f;   // 1/3
    const float PC2 = 0.2581988897471611f;   // 1/sqrt(15)
    const float colscale = (lo == 0) ? PC0 * 0.25f
                         : (lo == 1) ? PC1 * 0.25f
                         : (lo == 2) ? PC2 * 0.25f : 0.0f;
    const int col = (lo < 3) ? lo : 0;
    v2f b2[4];
#pragma unroll
    for (int c = 0; c < 4; ++c) {
        const int k0 = 4 * c + (hi ? 2 : 0);
        const int k1 = 4 * c + (hi ? 3 : 1);
        b2[c].x = W2[k0 * 3 + col] * colscale;
        b2[c].y = W2[k1 * 3 + col] * colscale;
    }

    float partial = 0.0f;

    for (int it = 0; it < n_iter; ++it) {
        const int  t  = wid + it * stride;
        const bool ok = (t < tiles);
        const int  tt = ok ? t : 0;
        const int  e0 = tt * 32 + lane;          // one edge per lane
        const bool evalid = (e0 < E);
        const int  e  = evalid ? e0 : (E - 1);

        // prefetch next iteration's streamed data into L2
        if (it + 1 < n_iter) {
            const int tn = wid + (it + 1) * stride;
            if (tn < tiles) {
                const int en = tn * 32 + lane;
                if (en < E) {
                    __builtin_prefetch(&ev[en * 3], 0, 1);
                    __builtin_prefetch(&esrc[en], 0, 1);
                }
            }
        }

        // ---- per-edge scalar math (one edge per lane, no duplication) ----
        const float x = ev[e * 3 + 0];
        const float y = ev[e * 3 + 1];
        const float z = ev[e * 3 + 2];
        const float r = sqrtf(x * x + y * y + z * z);
        const float inv = 1.0f / fmaxf(r, 1e-12f);
        const float ux = x * inv, uy = y * inv, uz = z * inv;

        // radial basis: zero-fill row, then scatter the <=2 live basis values.
        float* embrow = &lds_emb[wv][lane * EMB_PITCH];
        *(v4f*)(embrow + 0) = (v4f)0.0f;
        *(v4f*)(embrow + 4) = (v4f)0.0f;
        *(v4f*)(embrow + 8) = (v4f)0.0f;
        {
            const float CE = 1.14136f * 7.38905609893065f;   // 1.14136 * e^2
            const float s  = 11.0f * r;
            const int   ii = (int)floorf(s);
#pragma unroll
            for (int q = 0; q < 2; ++q) {
                const int k = ii - 1 + q;            // the only possibly-live bases
                if (k >= 0 && k <= 9) {
                    const float t1 = s - (float)k;          // diff + 1  (>= 0)
                    const float t2 = (float)(k + 2) - s;    // 1 - diff  (> 0)
                    // t1==0 -> rcp=+inf -> exp(-inf)=0, matching the cutoff
                    embrow[k] = CE * __expf(-(__builtin_amdgcn_rcpf(t1) +
                                              __builtin_amdgcn_rcpf(t2)));
                }
            }
        }

        // spherical harmonics (component-normalized, lmax=2) + gathered dot
        const float s3 = 1.7320508075688772f;
        const float s5 = 2.23606797749979f;
        const float x2 = ux * ux, y2 = uy * uy, z2 = uz * uz;
        const float sh4 = s5 * s3 * ux * uz;
        const float sh5 = s5 * s3 * ux * uy;
        const float sh6 = s5 * (y2 - 0.5f * (x2 + z2));
        const float sh7 = s5 * s3 * uy * uz;
        const float sh8 = s5 * 0.5f * s3 * (z2 - x2);

        const int src = esrc[e];
        const float* f = nf + src * 9;
        float d0 = f[0];
        float d1 = f[1] * (s3 * ux) + f[2] * (s3 * uy) + f[3] * (s3 * uz);
        float d2 = f[4] * sh4 + f[5] * sh5 + f[6] * sh6 + f[7] * sh7 + f[8] * sh8;
        if (!evalid) { d0 = 0.0f; d1 = 0.0f; d2 = 0.0f; }
        {
            v4f dv; dv.x = d0; dv.y = d1; dv.z = d2; dv.w = 0.0f;
            *(v4f*)&lds_d[wv][lane * 4] = dv;
        }
        __syncthreads();

        // ---- GEMM1 for both 16-edge halves ----
        // A fragment = one aligned b64 LDS load straight into an even VGPR pair.
        v8f c1[2];
#pragma unroll
        for (int p = 0; p < 2; ++p) {
            const float* arow = &lds_emb[wv][(p * 16 + lo) * EMB_PITCH];
            v8f acc = {};
#pragma unroll
            for (int c = 0; c < 3; ++c) {
                const v2f a = *(const v2f*)(arow + 4 * c + (hi ? 2 : 0));
                acc = __builtin_amdgcn_wmma_f32_16x16x4_f32(
                          false, a, false, b1[c], (short)0, acc, false, false);
            }
            c1[p] = acc;
        }

        // ReLU * sqrt(2); stage h tiles to LDS (C-layout -> A-layout transpose)
        const float s2 = 1.4142135623730951f;
#pragma unroll
        for (int p = 0; p < 2; ++p)
#pragma unroll
            for (int j = 0; j < 8; ++j) {
                const int m = j + (hi ? 8 : 0);
                lds_h[wv][p][m * H_PITCH + lo] = fmaxf(c1[p][j], 0.0f) * s2;
            }
        __syncthreads();

        // ---- GEMM2 for both halves ----
#pragma unroll
        for (int p = 0; p < 2; ++p) {
            const float* arow = &lds_h[wv][p][lo * H_PITCH];
            v8f c2 = {};
#pragma unroll
            for (int c = 0; c < 4; ++c) {
                const v2f a = *(const v2f*)(arow + 4 * c + (hi ? 2 : 0));
                c2 = __builtin_amdgcn_wmma_f32_16x16x4_f32(
                         false, a, false, b2[c], (short)0, c2, false, false);
            }
            // lane holds w'[m][lo]; dot against dots[m][lo], o = lo < 3
            if (ok && lo < 3) {
#pragma unroll
                for (int j = 0; j < 8; ++j) {
                    const int m = p * 16 + j + (hi ? 8 : 0);
                    partial += c2[j] * lds_d[wv][m * 4 + lo];
                }
            }
        }
        __syncthreads();   // protect per-wave LDS reuse next iteration
    }

    // ---- reduction: wave -> block ----
#pragma unroll
    for (int off = 16; off >= 1; off >>= 1)
        partial += __shfl_xor(partial, off, 32);
    if (lane == 0) red[wv] = partial;
    __syncthreads();
    if (tid == 0) {
        float s = 0.0f;
#pragma unroll
        for (int w = 0; w < WAVES_PER_BLOCK; ++w) s += red[w];
        partials[blockIdx.x] = s;
    }
}

__global__ void __launch_bounds__(256)
tsb_reduce_kernel(const float* __restrict__ partials, int n, float* __restrict__ out)
{
    __shared__ float red[256];
    float s = 0.0f;
    for (int i = threadIdx.x; i < n; i += 256) s += partials[i];
    red[threadIdx.x] = s;
    __syncthreads();
    for (int off = 128; off >= 1; off >>= 1) {
        if ((int)threadIdx.x < off) red[threadIdx.x] += red[threadIdx.x + off];
        __syncthreads();
    }
    if (threadIdx.x == 0)
        out[0] = red[0] * 0.31622776601683794f;   // 1/sqrt(NUM_NEIGHBORS)
}

extern "C" void kernel_launch(void* const* d_in, const int* in_sizes, int n_in,
                              void* d_out, int out_size, void* d_ws, size_t ws_size,
                              hipStream_t stream)
{
    const float* nf   = (const float*)d_in[0];   // node_feats [N,9]
    const float* ev   = (const float*)d_in[1];   // edge_vec   [E,3]
    const float* W1   = (const float*)d_in[2];   // [10,16]
    const float* W2   = (const float*)d_in[3];   // [16,3]
    const int*   esrc = (const int*)d_in[4];     // [E]
    // d_in[5] (edge_dst) is dead: segment_sum followed by .sum() == plain sum.

    const int E     = in_sizes[4];
    const int tiles = (E + 31) / 32;             // 32-edge tiles

    int nblocks = 625;                            // 625*8 waves = 5000 tiles/iter
    const size_t need = (size_t)nblocks * sizeof(float);
    if (ws_size < need) {
        nblocks = (int)(ws_size / sizeof(float));
        if (nblocks < 1) nblocks = 1;
    }
    const int stride = nblocks * WAVES_PER_BLOCK;
    const int n_iter = (tiles + stride - 1) / stride;

    float* partials = (float*)d_ws;
    tsb_edge_kernel<<<nblocks, BLOCK_THREADS, 0, stream>>>(
        nf, ev, W1, W2, esrc, partials, E, tiles, n_iter, stride);
    tsb_reduce_kernel<<<1, 256, 0, stream>>>(partials, nblocks, (float*)d_out);
}